// SparseLookupTable_21328807592432
// MI455X (gfx1250) — compile-verified
//
#include <hip/hip_runtime.h>
#include <stdint.h>

// SparseLookupTable on MI455X (gfx1250):
//   out[b,o] = sum_k lut[inp[b,k], wgt[o,k]],  B=512, O=1024, K=128, lut 256x256 f32.
// Gather-bound op (no multiply structure -> WMMA strictly worse: one-hot matmul
// operand staging alone exceeds direct-gather LDS traffic). CDNA5 path: stage
// the whole 256KB LUT in the 320KB WGP LDS via global_load_async_to_lds_b128
// (ASYNCcnt), then register-blocked LDS gather-accumulate.

#define LUT_STRIDE 260          // dwords per padded LUT row: 260*4=1040B (16B aligned),
                                // bank = (4*row + col) % 64 -> decorrelates rows
#define TILE_B 64
#define TILE_O 32
#define THREADS 256

#if __has_builtin(__builtin_amdgcn_global_load_async_to_lds_b128)
#define HAVE_ASYNC_LDS 1
#else
#define HAVE_ASYNC_LDS 0
#endif

typedef int v4i __attribute__((ext_vector_type(4)));
typedef __attribute__((address_space(1))) v4i gv4i;   // global
typedef __attribute__((address_space(3))) v4i lv4i;   // LDS

__global__ __launch_bounds__(THREADS)
void sparse_lut_kernel(const int* __restrict__ inp,    // (512,128) i32
                       const int* __restrict__ wgt,    // (1024,128) i32
                       const float* __restrict__ lut,  // (256,256) f32
                       float* __restrict__ out)        // (512,1024) f32
{
    extern __shared__ unsigned char smem[];
    float* lut_s = (float*)smem;                                   // 256 * 260 dwords
    unsigned char* inpT = smem + 256 * LUT_STRIDE * 4;             // TILE_B * 128 u8
    unsigned char* wgtT = inpT + TILE_B * 128;                     // TILE_O * 128 u8

    const int tid = threadIdx.x;
    const int b0  = blockIdx.y * TILE_B;
    const int o0  = blockIdx.x * TILE_O;

    // ---- stage LUT (256x256 f32 = 16384 16B chunks, 64 per row) into padded
    //      LDS rows. Constant trip count: no divergent-loop exec masking. ----
#pragma unroll 4
    for (int it = 0; it < 16384 / THREADS; ++it) {     // 64 iterations
        const int c   = tid + it * THREADS;            // chunk id, wave-contiguous
        const int row = c >> 6;                        // 64 chunks per row
        const int pos = (c & 63) << 2;                 // float offset within row
        const float* g = lut + row * 256 + pos;        // 16B aligned
        float* l = lut_s + row * LUT_STRIDE + pos;     // 16B aligned (1040B rows)
#if HAVE_ASYNC_LDS
        __builtin_amdgcn_global_load_async_to_lds_b128(
            (gv4i*)g, (lv4i*)l, /*imm offset*/0, /*cpol*/0);
#else
        *(float4*)l = *(const float4*)g;
#endif
    }

    // ---- stage index tiles, clamped to [0,255], compressed to u8 ----
    {
        const int* g = inp + b0 * 128;                 // contiguous 64x128 tile
#pragma unroll 4
        for (int it = 0; it < (TILE_B * 128) / THREADS; ++it) {   // 32 iterations
            const int e = tid + it * THREADS;
            int v = g[e];
            v = v < 0 ? 0 : (v > 255 ? 255 : v);
            inpT[e] = (unsigned char)v;
        }
    }
    {
        const int* g = wgt + o0 * 128;                 // contiguous 32x128 tile
#pragma unroll 4
        for (int it = 0; it < (TILE_O * 128) / THREADS; ++it) {   // 16 iterations
            const int e = tid + it * THREADS;
            int v = g[e];
            v = v < 0 ? 0 : (v > 255 ? 255 : v);
            wgtT[e] = (unsigned char)v;
        }
    }

#if HAVE_ASYNC_LDS
#if __has_builtin(__builtin_amdgcn_s_wait_asynccnt)
    __builtin_amdgcn_s_wait_asynccnt(0);
#else
    asm volatile("s_wait_asynccnt 0" ::: "memory");
#endif
#endif
    __syncthreads();

    // ---- register-blocked gather-accumulate: 4 b's x 2 o's per thread.
    //      Lanes 0..15 carry distinct o's -> randomized LDS banks; padded
    //      LUT rows decorrelate the two b half-waves. ----
    const int og = tid & 15;      // 16 o-groups * 2 o's = 32
    const int bg = tid >> 4;      // 16 b-groups * 4 b's = 64
    const int bb = bg * 4;
    const int oo = og * 2;

    const unsigned char* ip = inpT + bb * 128;
    const unsigned char* wp = wgtT + oo * 128;

    float acc00 = 0.f, acc01 = 0.f;
    float acc10 = 0.f, acc11 = 0.f;
    float acc20 = 0.f, acc21 = 0.f;
    float acc30 = 0.f, acc31 = 0.f;

#pragma unroll 2
    for (int kk = 0; kk < 128; kk += 4) {
        // 4 packed row-index bytes per b, 4 packed col-index bytes per o
        const uint32_t rv0 = *(const uint32_t*)(ip + 0 * 128 + kk);
        const uint32_t rv1 = *(const uint32_t*)(ip + 1 * 128 + kk);
        const uint32_t rv2 = *(const uint32_t*)(ip + 2 * 128 + kk);
        const uint32_t rv3 = *(const uint32_t*)(ip + 3 * 128 + kk);
        const uint32_t cv0 = *(const uint32_t*)(wp + 0 * 128 + kk);
        const uint32_t cv1 = *(const uint32_t*)(wp + 1 * 128 + kk);

#pragma unroll
        for (int s = 0; s < 32; s += 8) {
            const uint32_t r0 = ((rv0 >> s) & 255u) * LUT_STRIDE;
            const uint32_t r1 = ((rv1 >> s) & 255u) * LUT_STRIDE;
            const uint32_t r2 = ((rv2 >> s) & 255u) * LUT_STRIDE;
            const uint32_t r3 = ((rv3 >> s) & 255u) * LUT_STRIDE;
            const uint32_t c0 = (cv0 >> s) & 255u;
            const uint32_t c1 = (cv1 >> s) & 255u;

            acc00 += lut_s[r0 + c0];  acc01 += lut_s[r0 + c1];
            acc10 += lut_s[r1 + c0];  acc11 += lut_s[r1 + c1];
            acc20 += lut_s[r2 + c0];  acc21 += lut_s[r2 + c1];
            acc30 += lut_s[r3 + c0];  acc31 += lut_s[r3 + c1];
        }
    }

    // ---- coalesced float2 stores ----
    const int obase = o0 + oo;
    *(float2*)(out + (size_t)(b0 + bb + 0) * 1024 + obase) = make_float2(acc00, acc01);
    *(float2*)(out + (size_t)(b0 + bb + 1) * 1024 + obase) = make_float2(acc10, acc11);
    *(float2*)(out + (size_t)(b0 + bb + 2) * 1024 + obase) = make_float2(acc20, acc21);
    *(float2*)(out + (size_t)(b0 + bb + 3) * 1024 + obase) = make_float2(acc30, acc31);
}

extern "C" void kernel_launch(void* const* d_in, const int* in_sizes, int n_in,
                              void* d_out, int out_size, void* d_ws, size_t ws_size,
                              hipStream_t stream) {
    (void)in_sizes; (void)n_in; (void)out_size; (void)d_ws; (void)ws_size;
    const int*   inp = (const int*)d_in[0];    // (512,32,4) int32
    const int*   wgt = (const int*)d_in[1];    // (1024,32,4) int32
    const float* lut = (const float*)d_in[2];  // (256,256) float32
    float*       out = (float*)d_out;          // (512,1024) float32

    const dim3 grid(1024 / TILE_O, 512 / TILE_B);  // 32 x 8 = 256 blocks
    const size_t smem = (size_t)256 * LUT_STRIDE * 4   // 266240 B padded LUT
                      + (size_t)TILE_B * 128           //   8192 B inp tile (u8)
                      + (size_t)TILE_O * 128;          //   4096 B wgt tile (u8)
    sparse_lut_kernel<<<grid, THREADS, smem, stream>>>(inp, wgt, lut, out);
}